// VQ_VAE_72619307040979
// MI455X (gfx1250) — compile-verified
//
#include <hip/hip_runtime.h>
#include <hip/hip_bf16.h>

#define BTOT 131072
#define NE   2048
#define ZD   64
#define XD   5

typedef float v2f __attribute__((ext_vector_type(2)));
typedef float v8f __attribute__((ext_vector_type(8)));

__device__ __forceinline__ void stage(float* dst, const float* __restrict__ src,
                                      int n, int tid, int nthr) {
    for (int i = tid; i < n; i += nthr) dst[i] = src[i];
}

// ---------------- kernel 0: ||E_j||^2 ----------------
__global__ void __launch_bounds__(256)
vq_esq_kernel(const float* __restrict__ E, float* __restrict__ Esq) {
    int c = blockIdx.x * 256 + threadIdx.x;
    if (c < NE) {
        const float* row = E + (size_t)c * ZD;
        float s = 0.f;
        #pragma unroll
        for (int k = 0; k < ZD; ++k) s += row[k] * row[k];
        Esq[c] = s;
    }
}

// ---------------- kernel 1: encoder MLP ----------------
__global__ void __launch_bounds__(256)
vq_encode_kernel(const float* __restrict__ x,
                 const float* __restrict__ W1, const float* __restrict__ b1,
                 const float* __restrict__ W2, const float* __restrict__ b2,
                 const float* __restrict__ W3, const float* __restrict__ b3,
                 const float* __restrict__ W4, const float* __restrict__ b4,
                 float* __restrict__ z_e, float* __restrict__ zsq) {
    __shared__ float s[2256];
    float* sW1 = s;          // 5*16 = 80
    float* sb1 = s + 80;     // 16
    float* sW2 = s + 96;     // 16*32 = 512
    float* sb2 = s + 608;    // 32
    float* sW3 = s + 640;    // 32*16 = 512
    float* sb3 = s + 1152;   // 16
    float* sW4 = s + 1168;   // 16*64 = 1024
    float* sb4 = s + 2192;   // 64
    const int t = threadIdx.x;
    stage(sW1, W1, 80, t, 256);   stage(sb1, b1, 16, t, 256);
    stage(sW2, W2, 512, t, 256);  stage(sb2, b2, 32, t, 256);
    stage(sW3, W3, 512, t, 256);  stage(sb3, b3, 16, t, 256);
    stage(sW4, W4, 1024, t, 256); stage(sb4, b4, 64, t, 256);
    __syncthreads();

    const int row = blockIdx.x * 256 + threadIdx.x;
    float xi[XD];
    #pragma unroll
    for (int i = 0; i < XD; ++i) xi[i] = x[(size_t)row * XD + i];

    float h1[16];
    #pragma unroll
    for (int j = 0; j < 16; ++j) {
        float a = sb1[j];
        #pragma unroll
        for (int i = 0; i < XD; ++i) a += xi[i] * sW1[i * 16 + j];
        h1[j] = fmaxf(a, 0.f);
    }
    float h2[32];
    #pragma unroll
    for (int j = 0; j < 32; ++j) {
        float a = sb2[j];
        #pragma unroll
        for (int i = 0; i < 16; ++i) a += h1[i] * sW2[i * 32 + j];
        h2[j] = fmaxf(a, 0.f);
    }
    float h3[16];
    #pragma unroll
    for (int j = 0; j < 16; ++j) {
        float a = sb3[j];
        #pragma unroll
        for (int i = 0; i < 32; ++i) a += h2[i] * sW3[i * 16 + j];
        h3[j] = fmaxf(a, 0.f);
    }
    float ss = 0.f;
    #pragma unroll
    for (int j = 0; j < ZD; ++j) {
        float a = sb4[j];
        #pragma unroll
        for (int i = 0; i < 16; ++i) a += h3[i] * sW4[i * ZD + j];
        z_e[(size_t)row * ZD + j] = a;
        ss += a * a;
    }
    zsq[row] = ss;
}

// ---------------- kernel 2: WMMA nearest-codebook search ----------------
// Each wave owns a 16-row tile of z (A operand, resident in 32 VGPRs) and
// sweeps 128 N-tiles of E. score = ||E_n||^2 - 2 z.E_n (monotone in d^2).
__global__ void __launch_bounds__(128)
vq_quant_kernel(const float* __restrict__ z_e, const float* __restrict__ zsq,
                const float* __restrict__ E, const float* __restrict__ Esq,
                int* __restrict__ idx, float* __restrict__ embP) {
    __shared__ float sEsq[NE];
    __shared__ float sEmbed;
    for (int i = threadIdx.x; i < NE; i += 128) sEsq[i] = Esq[i];
    if (threadIdx.x == 0) sEmbed = 0.f;
    __syncthreads();

    const int wave = threadIdx.x >> 5;
    const int lane = threadIdx.x & 31;
    const int m    = lane & 15;
    const int hi   = lane >> 4;                 // K-half select (A/B layout)
    const int rowBase = (blockIdx.x * 4 + wave) * 16;

    // A tile: 16x64 f32 -> 16 chunks of 16x4; lane m holds row m, K pair per half
    v2f a[16];
    #pragma unroll
    for (int kk = 0; kk < 16; ++kk)
        a[kk] = *(const v2f*)(z_e + (size_t)(rowBase + m) * ZD + kk * 4 + hi * 2);

    float best[8];
    int   bidx[8];
    #pragma unroll
    for (int v = 0; v < 8; ++v) { best[v] = 3.4e38f; bidx[v] = 0; }

    for (int nt = 0; nt < NE / 16; ++nt) {
        const int n = nt * 16 + m;
        // B tile: 64x16 (B[k][n] = E[n][k]) -> lane-local float2 pairs
        v2f bt[16];
        #pragma unroll
        for (int kk = 0; kk < 16; ++kk)
            bt[kk] = *(const v2f*)(E + (size_t)n * ZD + kk * 4 + hi * 2);

        v8f acc = {0.f, 0.f, 0.f, 0.f, 0.f, 0.f, 0.f, 0.f};
        #pragma unroll
        for (int kk = 0; kk < 16; ++kk)
            acc = __builtin_amdgcn_wmma_f32_16x16x4_f32(
                false, a[kk], false, bt[kk], (short)0, acc, false, false);

        const float esq = sEsq[n];
        #pragma unroll
        for (int v = 0; v < 8; ++v) {
            float sc = esq - 2.0f * acc[v];   // D[v][lane]: M = v + hi*8, N = n
            if (sc < best[v]) { best[v] = sc; bidx[v] = n; }
        }
    }

    // reduce over the 16 lanes that share each row; tie-break = smallest index
    #pragma unroll
    for (int v = 0; v < 8; ++v) {
        #pragma unroll
        for (int off = 1; off < 16; off <<= 1) {
            float ob = __shfl_xor(best[v], off, 32);
            int   oi = __shfl_xor(bidx[v], off, 32);
            if (ob < best[v] || (ob == best[v] && oi < bidx[v])) {
                best[v] = ob; bidx[v] = oi;
            }
        }
    }

    if (m == 0) {
        float sum = 0.f;
        #pragma unroll
        for (int v = 0; v < 8; ++v) {
            const int row = rowBase + hi * 8 + v;
            idx[row] = bidx[v];
            sum += zsq[row] + best[v];        // = min d^2 for this row
        }
        atomicAdd(&sEmbed, sum);
    }
    __syncthreads();
    if (threadIdx.x == 0) embP[blockIdx.x] = 2.0f * sEmbed;  // embed + commit (BETA=1)
}

// ---------------- kernel 3: decoder MLP + recon L2 ----------------
__global__ void __launch_bounds__(256)
vq_decode_kernel(const float* __restrict__ x, const int* __restrict__ idx,
                 const float* __restrict__ E,
                 const float* __restrict__ W5, const float* __restrict__ b5,
                 const float* __restrict__ W6, const float* __restrict__ b6,
                 const float* __restrict__ W7, const float* __restrict__ b7,
                 const float* __restrict__ W8, const float* __restrict__ b8,
                 float* __restrict__ l2P) {
    __shared__ float s[2197];
    __shared__ float sL2;
    float* sW5 = s;          // 64*16 = 1024
    float* sb5 = s + 1024;   // 16
    float* sW6 = s + 1040;   // 16*32 = 512
    float* sb6 = s + 1552;   // 32
    float* sW7 = s + 1584;   // 32*16 = 512
    float* sb7 = s + 2096;   // 16
    float* sW8 = s + 2112;   // 16*5 = 80
    float* sb8 = s + 2192;   // 5
    const int t = threadIdx.x;
    stage(sW5, W5, 1024, t, 256); stage(sb5, b5, 16, t, 256);
    stage(sW6, W6, 512, t, 256);  stage(sb6, b6, 32, t, 256);
    stage(sW7, W7, 512, t, 256);  stage(sb7, b7, 16, t, 256);
    stage(sW8, W8, 80, t, 256);   stage(sb8, b8, 5, t, 256);
    if (t == 0) sL2 = 0.f;
    __syncthreads();

    const int row = blockIdx.x * 256 + threadIdx.x;
    const float* zq = E + (size_t)idx[row] * ZD;
    float zr[ZD];
    #pragma unroll
    for (int i = 0; i < ZD; ++i) zr[i] = zq[i];

    float h1[16];
    #pragma unroll
    for (int j = 0; j < 16; ++j) {
        float a = sb5[j];
        #pragma unroll
        for (int i = 0; i < ZD; ++i) a += zr[i] * sW5[i * 16 + j];
        h1[j] = fmaxf(a, 0.f);
    }
    float h2[32];
    #pragma unroll
    for (int j = 0; j < 32; ++j) {
        float a = sb6[j];
        #pragma unroll
        for (int i = 0; i < 16; ++i) a += h1[i] * sW6[i * 32 + j];
        h2[j] = fmaxf(a, 0.f);
    }
    float h3[16];
    #pragma unroll
    for (int j = 0; j < 16; ++j) {
        float a = sb7[j];
        #pragma unroll
        for (int i = 0; i < 32; ++i) a += h2[i] * sW7[i * 16 + j];
        h3[j] = fmaxf(a, 0.f);
    }
    float l2 = 0.f;
    #pragma unroll
    for (int j = 0; j < XD; ++j) {
        float a = sb8[j];
        #pragma unroll
        for (int i = 0; i < 16; ++i) a += h3[i] * sW8[i * XD + j];
        float d = x[(size_t)row * XD + j] - a;
        l2 += d * d;
    }
    atomicAdd(&sL2, l2);
    __syncthreads();
    if (t == 0) l2P[blockIdx.x] = sL2;
}

// ---------------- kernel 4: final reduction -> 3 scalars ----------------
__global__ void __launch_bounds__(256)
vq_final_kernel(const float* __restrict__ embP, int nEmb,
                const float* __restrict__ l2P, int nL2,
                float* __restrict__ out) {
    __shared__ float sE[256], sL[256];
    float e = 0.f, l = 0.f;
    for (int i = threadIdx.x; i < nEmb; i += 256) e += embP[i];
    for (int i = threadIdx.x; i < nL2;  i += 256) l += l2P[i];
    sE[threadIdx.x] = e; sL[threadIdx.x] = l;
    __syncthreads();
    for (int st = 128; st > 0; st >>= 1) {
        if (threadIdx.x < st) {
            sE[threadIdx.x] += sE[threadIdx.x + st];
            sL[threadIdx.x] += sL[threadIdx.x + st];
        }
        __syncthreads();
    }
    if (threadIdx.x == 0) {
        const float C = -0.5f * (float)XD * logf(2.0f * 3.14159265358979323f / 10.0f);
        float neg_pxz = 0.5f * 10.0f * sL[0] - C;   // -pxz
        out[0] = neg_pxz + sE[0];                   // loss
        out[1] = neg_pxz;
        out[2] = sE[0];                             // embedding loss
    }
}

extern "C" void kernel_launch(void* const* d_in, const int* in_sizes, int n_in,
                              void* d_out, int out_size, void* d_ws, size_t ws_size,
                              hipStream_t stream) {
    (void)in_sizes; (void)n_in; (void)out_size; (void)ws_size;
    const float* x  = (const float*)d_in[0];
    const float* W1 = (const float*)d_in[1];  const float* b1 = (const float*)d_in[2];
    const float* W2 = (const float*)d_in[3];  const float* b2 = (const float*)d_in[4];
    const float* W3 = (const float*)d_in[5];  const float* b3 = (const float*)d_in[6];
    const float* W4 = (const float*)d_in[7];  const float* b4 = (const float*)d_in[8];
    const float* W5 = (const float*)d_in[9];  const float* b5 = (const float*)d_in[10];
    const float* W6 = (const float*)d_in[11]; const float* b6 = (const float*)d_in[12];
    const float* W7 = (const float*)d_in[13]; const float* b7 = (const float*)d_in[14];
    const float* W8 = (const float*)d_in[15]; const float* b8 = (const float*)d_in[16];
    const float* E  = (const float*)d_in[17];
    float* out = (float*)d_out;

    // workspace layout (floats)
    char* ws = (char*)d_ws;
    float* z_e  = (float*)ws;                                  // B*64
    float* zsq  = (float*)(ws + (size_t)BTOT * ZD * 4);        // B
    int*   idx  = (int*)  (ws + (size_t)BTOT * (ZD + 1) * 4);  // B
    float* Esq  = (float*)(ws + (size_t)BTOT * (ZD + 2) * 4);  // NE
    float* embP = Esq + NE;                                    // 2048
    float* l2P  = embP + 2048;                                 // 512

    vq_esq_kernel<<<(NE + 255) / 256, 256, 0, stream>>>(E, Esq);
    vq_encode_kernel<<<BTOT / 256, 256, 0, stream>>>(
        x, W1, b1, W2, b2, W3, b3, W4, b4, z_e, zsq);
    vq_quant_kernel<<<BTOT / 64, 128, 0, stream>>>(z_e, zsq, E, Esq, idx, embP);
    vq_decode_kernel<<<BTOT / 256, 256, 0, stream>>>(
        x, idx, E, W5, b5, W6, b6, W7, b7, W8, b8, l2P);
    vq_final_kernel<<<1, 256, 0, stream>>>(embP, BTOT / 64, l2P, BTOT / 256, out);
}